// BEVEncoderLayer_28467043238169
// MI455X (gfx1250) — compile-verified
//
#include <hip/hip_runtime.h>
#include <hip/hip_bf16.h>

typedef __attribute__((ext_vector_type(2))) float v2f;
typedef __attribute__((ext_vector_type(8))) float v8f;
typedef int v4i_vs __attribute__((__vector_size__(16)));   // matches async builtin param

#define LQ     10000
#define CDIM   256
#define NHEAD  8
#define HD     32
#define NPTS   8
#define NCAMS  6
#define DANCH  4
#define H_BEV  100
#define W_BEV  100
#define H_IMG  80
#define W_IMG  120
#define DFF    512
#define HW_IMG (H_IMG * W_IMG)   // 9600
#define MVAL   (NCAMS * HW_IMG)  // 57600
#define KC     128               // K-chunk staged in LDS (32 KB)

#if __has_builtin(__builtin_amdgcn_global_load_async_to_lds_b128) && \
    __has_builtin(__builtin_amdgcn_s_wait_asynccnt)
#define HAVE_ASYNC_LDS 1
#else
#define HAVE_ASYNC_LDS 0
#endif

// ---------------------------------------------------------------------------
// GEMM: out[M,N] = A[M,K] @ W[K,N] (+bias) (opt ReLU), fp32 via
// V_WMMA_F32_16X16X4_F32. Workgroup = 8 waves; each wave owns a 16(M)x64(N)
// tile; all 8 waves share one 64-column B panel staged through LDS in
// KC-row chunks (32 KB), cutting global-side B traffic ~8x and turning it
// into b128 transfers (async global->LDS when the toolchain exposes it).
// Requires M%16==0, K%KC==0, N%64==0 (true for all call sites here).
// ---------------------------------------------------------------------------
__global__ __launch_bounds__(256) void gemm_wmma_f32(
    const float* __restrict__ A, const float* __restrict__ W,
    const float* __restrict__ bias, float* __restrict__ out,
    int M, int K, int N, int relu)
{
    __shared__ float sB[KC * 64];

    const int tid   = threadIdx.x;
    const int lane  = tid & 31;
    const int wave  = tid >> 5;
    const int mtile = M >> 4;
    const int mblocks = (mtile + 7) >> 3;
    const int mblk = blockIdx.x % mblocks;
    const int nblk = blockIdx.x / mblocks;

    const int mt = mblk * 8 + wave;
    const bool valid = (mt < mtile);
    const int m0 = mt << 4;
    const int n0 = nblk << 6;

    const int row = lane & 15;
    const int hi  = lane >> 4;

    v8f acc0 = {}, acc1 = {}, acc2 = {}, acc3 = {};
    const float* Arow = A + (size_t)(m0 + row) * K + hi * 2;

    for (int kc = 0; kc < K; kc += KC) {
        // ---- stage B chunk: rows [kc, kc+KC) x cols [n0, n0+64) ----------
        for (int i = tid; i < KC * 16; i += 256) {
            int r  = i >> 4;         // 0..KC-1
            int c4 = (i & 15) * 4;   // 0..60
            const float* src = W + (size_t)(kc + r) * N + n0 + c4;
#if HAVE_ASYNC_LDS
            __builtin_amdgcn_global_load_async_to_lds_b128(
                (v4i_vs*)src, (v4i_vs*)&sB[r * 64 + c4], 0, 0);
#else
            *(float4*)&sB[r * 64 + c4] = *(const float4*)src;
#endif
        }
#if HAVE_ASYNC_LDS
        __builtin_amdgcn_s_wait_asynccnt(0);
#endif
        __syncthreads();

        if (valid) {
            for (int k0 = 0; k0 < KC; k0 += 4) {
                v2f a = *(const v2f*)(Arow + kc + k0);
                const float* Wk = sB + (k0 + hi * 2) * 64 + row;
                v2f b0, b1, b2, b3;
                b0.x = Wk[0];  b0.y = Wk[64 + 0];
                b1.x = Wk[16]; b1.y = Wk[64 + 16];
                b2.x = Wk[32]; b2.y = Wk[64 + 32];
                b3.x = Wk[48]; b3.y = Wk[64 + 48];
                acc0 = __builtin_amdgcn_wmma_f32_16x16x4_f32(false, a, false, b0, (short)0, acc0, false, false);
                acc1 = __builtin_amdgcn_wmma_f32_16x16x4_f32(false, a, false, b1, (short)0, acc1, false, false);
                acc2 = __builtin_amdgcn_wmma_f32_16x16x4_f32(false, a, false, b2, (short)0, acc2, false, false);
                acc3 = __builtin_amdgcn_wmma_f32_16x16x4_f32(false, a, false, b3, (short)0, acc3, false, false);
            }
        }
        __syncthreads();
    }

    if (!valid) return;
    v8f accs[4] = {acc0, acc1, acc2, acc3};
#pragma unroll
    for (int t = 0; t < 4; ++t) {
        int col = n0 + t * 16 + row;
        float bv = bias ? bias[col] : 0.0f;
#pragma unroll
        for (int i = 0; i < 8; ++i) {
            int r = m0 + i + hi * 8;
            float v = accs[t][i] + bv;
            if (relu) v = fmaxf(v, 0.0f);
            out[(size_t)r * N + col] = v;
        }
    }
}

// ---------------------------------------------------------------------------
// Bilinear sample (grid_sample align_corners=False, zero padding). Channel
// stride is CDIM (value layout: pixel-major, channel = head*HD + lane).
// ---------------------------------------------------------------------------
__device__ __forceinline__ float bilin(const float* __restrict__ v, int H, int W,
                                       float x, float y, int ch)
{
    float x0f = floorf(x), y0f = floorf(y);
    int x0 = (int)x0f, y0 = (int)y0f;
    float lx = x - x0f, ly = y - y0f;
    float s = 0.0f;
#pragma unroll
    for (int dy = 0; dy < 2; ++dy) {
#pragma unroll
        for (int dx = 0; dx < 2; ++dx) {
            int ix = x0 + dx, iy = y0 + dy;
            if (ix >= 0 && ix < W && iy >= 0 && iy < H) {
                float wt = (dx ? lx : 1.0f - lx) * (dy ? ly : 1.0f - ly);
                s += wt * v[(size_t)(iy * W + ix) * CDIM + ch];
            }
        }
    }
    return s;
}

// ---------------------------------------------------------------------------
// Temporal self-attn deformable sampling. One block per query, one wave per
// head, lane = channel (HD == 32 == wave32).
// ---------------------------------------------------------------------------
__global__ __launch_bounds__(256) void tsa_sample_kernel(
    const float* __restrict__ val,    // (LQ, CDIM)
    const float* __restrict__ off,    // (LQ, 128)
    const float* __restrict__ logits, // (LQ, 64)
    const float* __restrict__ ref2d,  // (LQ, 2)
    float* __restrict__ outbuf)       // (LQ, CDIM)
{
    int q = blockIdx.x;
    int h = threadIdx.x >> 5;
    int lane = threadIdx.x & 31;

    const float* lg = logits + q * 64 + h * NPTS;
    float mx = -1e30f;
#pragma unroll
    for (int p = 0; p < NPTS; ++p) mx = fmaxf(mx, lg[p]);
    float w[NPTS], den = 0.0f;
#pragma unroll
    for (int p = 0; p < NPTS; ++p) { w[p] = __expf(lg[p] - mx); den += w[p]; }
    float rden = 1.0f / den;

    float rx = ref2d[q * 2 + 0], ry = ref2d[q * 2 + 1];
    const float* of = off + q * 128 + h * 16;
    int ch = h * HD + lane;
    float acc = 0.0f;
#pragma unroll
    for (int p = 0; p < NPTS; ++p) {
        float x = (rx + of[p * 2 + 0] * (1.0f / (float)W_BEV)) * (float)W_BEV - 0.5f;
        float y = (ry + of[p * 2 + 1] * (1.0f / (float)H_BEV)) * (float)H_BEV - 0.5f;
        acc += (w[p] * rden) * bilin(val, H_BEV, W_BEV, x, y, ch);
    }
    outbuf[(size_t)q * CDIM + ch] = acc;
}

// ---------------------------------------------------------------------------
// Spatial cross-attn deformable sampling, per-camera masked average.
// ---------------------------------------------------------------------------
__global__ __launch_bounds__(256) void sca_sample_kernel(
    const float* __restrict__ val,    // (NCAMS, HW_IMG, CDIM)
    const float* __restrict__ off,    // (LQ, 128)
    const float* __restrict__ logits, // (LQ, 64)
    const float* __restrict__ refcam, // (NCAMS, LQ, DANCH, 2)
    const int*   __restrict__ mask,   // (NCAMS, LQ, DANCH)
    float* __restrict__ outbuf)       // (LQ, CDIM)
{
    int q = blockIdx.x;
    int h = threadIdx.x >> 5;
    int lane = threadIdx.x & 31;

    const float* lg = logits + q * 64 + h * NPTS;
    float mx = -1e30f;
#pragma unroll
    for (int p = 0; p < NPTS; ++p) mx = fmaxf(mx, lg[p]);
    float w[NPTS], den = 0.0f;
#pragma unroll
    for (int p = 0; p < NPTS; ++p) { w[p] = __expf(lg[p] - mx); den += w[p]; }
    float rden = 1.0f / den;

    const float* of = off + q * 128 + h * 16;
    int ch = h * HD + lane;
    float slots = 0.0f, cnt = 0.0f;

    for (int cam = 0; cam < NCAMS; ++cam) {
        const int* mk = mask + (size_t)cam * LQ * DANCH + q * DANCH;
        int vis = mk[0] + mk[1] + mk[2] + mk[3];
        if (vis <= 0) continue;
        cnt += 1.0f;
        const float* rp = refcam + (size_t)cam * LQ * DANCH * 2 + q * DANCH * 2;
        const float* v  = val + (size_t)cam * HW_IMG * CDIM;
        float acc = 0.0f;
#pragma unroll
        for (int p = 0; p < NPTS; ++p) {
            int d = p & (DANCH - 1);   // reshape (NP//D, D): anchor = p % D
            float x = (rp[d * 2 + 0] + of[p * 2 + 0] * (1.0f / (float)W_IMG)) * (float)W_IMG - 0.5f;
            float y = (rp[d * 2 + 1] + of[p * 2 + 1] * (1.0f / (float)H_IMG)) * (float)H_IMG - 0.5f;
            acc += (w[p] * rden) * bilin(v, H_IMG, W_IMG, x, y, ch);
        }
        slots += acc;
    }
    outbuf[(size_t)q * CDIM + ch] = slots / fmaxf(cnt, 1.0f);
}

// ---------------------------------------------------------------------------
// out = LayerNorm(resid + x) * g + b, one block (256 threads) per row.
// ---------------------------------------------------------------------------
__global__ __launch_bounds__(256) void add_ln_kernel(
    const float* __restrict__ resid, const float* __restrict__ x,
    const float* __restrict__ g, const float* __restrict__ b,
    float* __restrict__ out)
{
    __shared__ float red[CDIM];
    int q = blockIdx.x, c = threadIdx.x;
    float y = resid[(size_t)q * CDIM + c] + x[(size_t)q * CDIM + c];

    red[c] = y;
    __syncthreads();
    for (int s = CDIM / 2; s > 0; s >>= 1) {
        if (c < s) red[c] += red[c + s];
        __syncthreads();
    }
    float mean = red[0] * (1.0f / (float)CDIM);
    __syncthreads();

    float d = y - mean;
    red[c] = d * d;
    __syncthreads();
    for (int s = CDIM / 2; s > 0; s >>= 1) {
        if (c < s) red[c] += red[c + s];
        __syncthreads();
    }
    float var = red[0] * (1.0f / (float)CDIM);
    float r = rsqrtf(var + 1e-5f);
    out[(size_t)q * CDIM + c] = d * r * g[c] + b[c];
}

// ---------------------------------------------------------------------------
extern "C" void kernel_launch(void* const* d_in, const int* in_sizes, int n_in,
                              void* d_out, int out_size, void* d_ws, size_t ws_size,
                              hipStream_t stream)
{
    const float* query   = (const float*)d_in[0];
    const float* prev    = (const float*)d_in[1];
    const float* ref2d   = (const float*)d_in[2];
    const float* refcam  = (const float*)d_in[3];
    const float* valimg  = (const float*)d_in[4];
    const int*   bevmask = (const int*)d_in[5];
    // params in dict order
    const float* tsa_off_w  = (const float*)d_in[6];   // (512,128) -> top 256 rows (qt tail is zeros)
    const float* tsa_off_b  = (const float*)d_in[7];
    const float* tsa_attn_w = (const float*)d_in[8];   // (512,64)  -> top 256 rows
    const float* tsa_attn_b = (const float*)d_in[9];
    const float* tsa_val_w  = (const float*)d_in[10];
    const float* tsa_val_b  = (const float*)d_in[11];
    const float* tsa_out_w  = (const float*)d_in[12];
    const float* tsa_out_b  = (const float*)d_in[13];
    const float* sca_off_w  = (const float*)d_in[14];
    const float* sca_off_b  = (const float*)d_in[15];
    const float* sca_attn_w = (const float*)d_in[16];
    const float* sca_attn_b = (const float*)d_in[17];
    const float* sca_val_w  = (const float*)d_in[18];
    const float* sca_val_b  = (const float*)d_in[19];
    const float* sca_out_w  = (const float*)d_in[20];
    const float* sca_out_b  = (const float*)d_in[21];
    const float* ffn_w1     = (const float*)d_in[22];
    const float* ffn_b1     = (const float*)d_in[23];
    const float* ffn_w2     = (const float*)d_in[24];
    const float* ffn_b2     = (const float*)d_in[25];
    const float* ln1_g = (const float*)d_in[26];
    const float* ln1_b = (const float*)d_in[27];
    const float* ln2_g = (const float*)d_in[28];
    const float* ln2_b = (const float*)d_in[29];
    const float* ln3_g = (const float*)d_in[30];
    const float* ln3_b = (const float*)d_in[31];

    float* ws = (float*)d_ws;
    size_t o = 0;
    float* tsa_val = ws + o; o += (size_t)LQ * CDIM;
    float* off_buf = ws + o; o += (size_t)LQ * 128;   // reused TSA then SCA
    float* att_buf = ws + o; o += (size_t)LQ * 64;    // reused TSA then SCA
    float* samp    = ws + o; o += (size_t)LQ * CDIM;  // deform output / slots
    float* proj    = ws + o; o += (size_t)LQ * CDIM;  // out-projection result
    float* q_ln1   = ws + o; o += (size_t)LQ * CDIM;
    float* q_ln2   = ws + o; o += (size_t)LQ * CDIM;
    float* ffn_h   = ws + o; o += (size_t)LQ * DFF;
    float* sca_val = ws + o; o += (size_t)MVAL * CDIM;
    (void)ws_size; (void)in_sizes; (void)n_in; (void)out_size;

    auto gemm = [&](const float* A, const float* W, const float* bias, float* out,
                    int M, int K, int N, int relu) {
        int mtile = M / 16;
        int mblocks = (mtile + 7) / 8;
        int blocks = mblocks * (N / 64);
        gemm_wmma_f32<<<blocks, 256, 0, stream>>>(A, W, bias, out, M, K, N, relu);
    };

    // ---- Phase 1: temporal self-attention -------------------------------
    gemm(prev,  tsa_val_w,  tsa_val_b,  tsa_val, LQ, CDIM, CDIM, 0);
    gemm(query, tsa_off_w,  tsa_off_b,  off_buf, LQ, CDIM, 128,  0);
    gemm(query, tsa_attn_w, tsa_attn_b, att_buf, LQ, CDIM, 64,   0);
    tsa_sample_kernel<<<LQ, 256, 0, stream>>>(tsa_val, off_buf, att_buf, ref2d, samp);
    gemm(samp, tsa_out_w, tsa_out_b, proj, LQ, CDIM, CDIM, 0);
    add_ln_kernel<<<LQ, 256, 0, stream>>>(query, proj, ln1_g, ln1_b, q_ln1);

    // ---- Phase 2: spatial cross-attention -------------------------------
    gemm(valimg, sca_val_w,  sca_val_b,  sca_val, MVAL, CDIM, CDIM, 0);
    gemm(q_ln1,  sca_off_w,  sca_off_b,  off_buf, LQ,   CDIM, 128,  0);
    gemm(q_ln1,  sca_attn_w, sca_attn_b, att_buf, LQ,   CDIM, 64,   0);
    sca_sample_kernel<<<LQ, 256, 0, stream>>>(sca_val, off_buf, att_buf, refcam, bevmask, samp);
    gemm(samp, sca_out_w, sca_out_b, proj, LQ, CDIM, CDIM, 0);
    add_ln_kernel<<<LQ, 256, 0, stream>>>(q_ln1, proj, ln2_g, ln2_b, q_ln2);

    // ---- Phase 3: FFN ---------------------------------------------------
    gemm(q_ln2, ffn_w1, ffn_b1, ffn_h, LQ, CDIM, DFF, 1);   // ReLU fused
    gemm(ffn_h, ffn_w2, ffn_b2, proj,  LQ, DFF,  CDIM, 0);
    add_ln_kernel<<<LQ, 256, 0, stream>>>(q_ln2, proj, ln3_g, ln3_b, (float*)d_out);
}